// MFELoss_6236292514426
// MI455X (gfx1250) — compile-verified
//
#include <hip/hip_runtime.h>
#include <cstdint>

// ---------------------------------------------------------------------------
// MFE loss: streaming softmax-derived reduction over 8.4M x 4 f32.
// Bandwidth-bound (~160 MiB @ 23.3 TB/s ~= 7us). CDNA5 async global->LDS
// loads (ASYNCcnt) in a depth-4 pipeline: preds (b128) + target (b32) per
// tile, 3 tiles in flight per wave at steady state (~25 MB aggregate across
// 16K waves > HBM latency-BW product).
// ---------------------------------------------------------------------------

#define BLK   256
#define DEPTH 4

__device__ __forceinline__ float waveSum(float v) {
    v += __shfl_xor(v, 16, 32);
    v += __shfl_xor(v, 8, 32);
    v += __shfl_xor(v, 4, 32);
    v += __shfl_xor(v, 2, 32);
    v += __shfl_xor(v, 1, 32);
    return v;
}

// Reduce 3 values across a 256-thread block (8 wave32s). Result valid in
// thread 0.
__device__ __forceinline__ void blockReduce3(float& a, float& b, float& c) {
    __shared__ float red[3][8];
    const int lane = threadIdx.x & 31;
    const int wid  = threadIdx.x >> 5;
    a = waveSum(a); b = waveSum(b); c = waveSum(c);
    if (lane == 0) { red[0][wid] = a; red[1][wid] = b; red[2][wid] = c; }
    __syncthreads();
    if (wid == 0) {
        const int nw = (int)(blockDim.x >> 5);
        float x = (lane < nw) ? red[0][lane] : 0.0f;
        float y = (lane < nw) ? red[1][lane] : 0.0f;
        float z = (lane < nw) ? red[2][lane] : 0.0f;
        a = waveSum(x); b = waveSum(y); c = waveSum(z);
    }
}

__device__ __forceinline__ void asyncLoadB128(unsigned ldsAddr, unsigned byteOff,
                                              unsigned long long base) {
    asm volatile("global_load_async_to_lds_b128 %0, %1, %2"
                 :: "v"(ldsAddr), "v"(byteOff), "s"(base) : "memory");
}
__device__ __forceinline__ void asyncLoadB32(unsigned ldsAddr, unsigned byteOff,
                                             unsigned long long base) {
    asm volatile("global_load_async_to_lds_b32 %0, %1, %2"
                 :: "v"(ldsAddr), "v"(byteOff), "s"(base) : "memory");
}

__global__ void __launch_bounds__(BLK)
mfe_main(const float4* __restrict__ preds,   // [n] rows of 4 f32 (16B)
         const int*    __restrict__ target,  // [n]
         const int*    __restrict__ oip,     // scalar others_idx
         float*        __restrict__ part,    // [3*gridDim.x] partials
         int n)
{
    __shared__ float4 tileP[DEPTH][BLK];   // 16 KB
    __shared__ int    tileT[DEPTH][BLK];   //  4 KB

    const int tid   = threadIdx.x;
    const int G     = gridDim.x;
    const int tiles = (n + BLK - 1) / BLK;
    const int oi    = oip[0];
    const unsigned long long baseP = (unsigned long long)preds;
    const unsigned long long baseT = (unsigned long long)target;
    const float L2E = 1.4426950408889634f;

    float fne = 0.0f, fpe = 0.0f, cnt = 0.0f;

    const int t0 = blockIdx.x;

    // Prime the pipeline: tiles t0 .. t0+(DEPTH-2)*G into buffers 0..DEPTH-2.
    #pragma unroll
    for (int k = 0; k < DEPTH - 1; ++k) {
        const int tk = t0 + k * G;
        if (tk < tiles) {
            const int row = tk * BLK + tid;
            if (row < n) {
                asyncLoadB128((unsigned)(uintptr_t)&tileP[k][tid],
                              (unsigned)row * 16u, baseP);
                asyncLoadB32((unsigned)(uintptr_t)&tileT[k][tid],
                             (unsigned)row * 4u, baseT);
            }
        }
    }

    int it = 0;
    for (int t = t0; t < tiles; t += G, ++it) {
        const int buf = it & (DEPTH - 1);
        const int tp  = t + (DEPTH - 1) * G;   // tile to prefetch

        if (tp < tiles) {
            // The buffer we are about to overwrite was ds_load-read last
            // iteration; async LDS writes are unordered vs DS reads, so
            // drain DS first (free: those reads fed the previous compute).
            asm volatile("s_wait_dscnt 0x0" ::: "memory");
            const int row2 = tp * BLK + tid;
            if (row2 < n) {
                const int pbuf = (it + DEPTH - 1) & (DEPTH - 1);
                asyncLoadB128((unsigned)(uintptr_t)&tileP[pbuf][tid],
                              (unsigned)row2 * 16u, baseP);
                asyncLoadB32((unsigned)(uintptr_t)&tileT[pbuf][tid],
                             (unsigned)row2 * 4u, baseT);
            }
            // 2 asyncs per tile, in-order completion: leaving 6 outstanding
            // means the oldest tile's pair (current buffer) has landed.
            asm volatile("s_wait_asynccnt 0x6" ::: "memory");
        } else {
            // Pipeline tail: drain everything.
            asm volatile("s_wait_asynccnt 0x0" ::: "memory");
        }

        const int row = t * BLK + tid;
        if (row < n) {
            const int    tv = tileT[buf][tid];  // ds_load_b32
            const float4 pr = tileP[buf][tid];  // ds_load_b128

            // Numerically-stable softmax over 4 logits.
            float m  = fmaxf(fmaxf(pr.x, pr.y), fmaxf(pr.z, pr.w));
            float e0 = exp2f((pr.x - m) * L2E);
            float e1 = exp2f((pr.y - m) * L2E);
            float e2 = exp2f((pr.z - m) * L2E);
            float e3 = exp2f((pr.w - m) * L2E);
            float den = (e0 + e1) + (e2 + e3);

            float inv = __builtin_amdgcn_rcpf(den);
            inv = inv * (2.0f - den * inv);     // one Newton step

            float eo = (oi == 0) ? e0 : (oi == 1) ? e1 : (oi == 2) ? e2 : e3;
            float po = eo * inv;
            float s  = den * inv;               // == probs.sum(axis=1)

            float d1 = s - po;
            float d2 = po - 1.0f;
            float fne_i = 0.5f * (d1 * d1 + d2 * d2);
            float fpe_i = po * po;              // 0.5*((-p)^2 + p^2)

            const bool is_o = (tv == oi);
            fne += is_o ? fne_i : 0.0f;
            fpe += is_o ? 0.0f  : fpe_i;
            cnt += is_o ? 1.0f  : 0.0f;
        }
    }

    blockReduce3(fne, fpe, cnt);
    if (tid == 0) {
        part[blockIdx.x]         = fne;
        part[G + blockIdx.x]     = fpe;
        part[2 * G + blockIdx.x] = cnt;
    }
}

__global__ void __launch_bounds__(256)
mfe_finalize(const float* __restrict__ part, int G,
             float* __restrict__ out, float total)
{
    float a = 0.0f, b = 0.0f, c = 0.0f;
    for (int i = threadIdx.x; i < G; i += 256) {
        a += part[i];
        b += part[G + i];
        c += part[2 * G + i];
    }
    blockReduce3(a, b, c);
    if (threadIdx.x == 0) {
        float fne_num = c;
        float fpe_num = total - c;
        float fne = (fne_num > 0.0f) ? a / fne_num : 0.0f;
        float fpe = (fpe_num > 0.0f) ? b / fpe_num : 0.0f;
        out[0] = fne + fpe;
    }
}

extern "C" void kernel_launch(void* const* d_in, const int* in_sizes, int n_in,
                              void* d_out, int out_size, void* d_ws, size_t ws_size,
                              hipStream_t stream)
{
    const float4* preds  = (const float4*)d_in[0];
    const int*    target = (const int*)d_in[1];
    const int*    oip    = (const int*)d_in[2];
    float*        out    = (float*)d_out;

    const int n = in_sizes[1];   // number of rows (target element count)

    int G = 2048;
    if (ws_size < (size_t)G * 3 * sizeof(float)) {
        G = (int)(ws_size / (3 * sizeof(float)));
        if (G < 1) G = 1;
    }
    float* part = (float*)d_ws;

    mfe_main<<<G, BLK, 0, stream>>>(preds, target, oip, part, n);
    mfe_finalize<<<1, 256, 0, stream>>>(part, G, out, (float)n);

    (void)n_in; (void)out_size;
}